// C4StandardTransformer_62380105007582
// MI455X (gfx1250) — compile-verified
//
#include <hip/hip_runtime.h>
#include <math.h>

// ---------------------------------------------------------------------------
// C4StandardTransformer on MI455X (gfx1250, wave32, WMMA)
//
//  * softmax over a size-1 axis == 1  => Q, K, scores, softmax are dead.
//  * attn = Xn * Wv^T * Wo^T = Xn * (Wo*Wv)^T  => precompute Wvo per expert.
//  * D = 16 matches the 16x16 WMMA tile; all GEMMs use V_WMMA_F32_16X16X4_F32
//    (full fp32 precision, K=16/64 via chained k-steps of 4).
//  * One wave32 per 16-row tile; single-wave workgroups (barriers are free,
//    waves fully independent). Gated expert sum accumulated in registers.
//  * silu via v_tanh_f32 (3 VALU), gate via 2x v_exp_f32 + 1x v_rcp_f32;
//    layernorm via v_rsq_f32. Trans ops batched to hide TRANS latency.
//  * All B-frags of a GEMM loaded in one clause; WMMAs round-robin over
//    independent accumulators.
// ---------------------------------------------------------------------------

#define NE 39
#define ND 16
#define NDFF 64
#define NB 32768
#define OPCODE_COL 6
#define LN_EPS 1e-5f
#define L2E20 28.853900817779268f   // 20*log2(e)
#define L2E10 14.426950408889634f   // 10*log2(e)

typedef __attribute__((ext_vector_type(2))) float v2f;
typedef __attribute__((ext_vector_type(8))) float v8f;

// A-frag / transposed-B-frag loader.
// WMMA f32 16x16x4 A layout: lane l, vgpr j  <- A[l&15][k0 + 2*(l>>4) + j].
// For B we need B[k][n] = W[n][k] (W row-major, out x in); under the
// symmetric B layout, lane l vgpr j <- B[k0+2*hi+j][l&15] = W[l&15][k0+2*hi+j]
// -- the SAME per-lane indexing as the A-frag.
__device__ __forceinline__ v2f frag16(const float* p, int ln, int hi, int k0, int stride) {
  const float* q = p + ln * stride + k0 + 2 * hi;
  return v2f{q[0], q[1]};
}

__device__ __forceinline__ v8f wmma4(v2f a, v2f b, v8f c) {
  // (neg_a, A, neg_b, B, c_mod, C, reuse_a, reuse_b)
  return __builtin_amdgcn_wmma_f32_16x16x4_f32(false, a, false, b, (short)0, c,
                                               false, false);
}

__device__ __forceinline__ float fast_rcp(float x) { return __builtin_amdgcn_rcpf(x); }
__device__ __forceinline__ float fast_rsq(float x) { return __builtin_amdgcn_rsqf(x); }
__device__ __forceinline__ float fast_exp2(float x) { return __builtin_amdgcn_exp2f(x); }

// tanh half of silu: given u = z/2, returns tanh(u) (v_tanh_f32 if available).
__device__ __forceinline__ float fast_tanh(float u) {
#if __has_builtin(__builtin_amdgcn_tanhf)
  return __builtin_amdgcn_tanhf(u);
#else
  // tanh(u) = 1 - 2/(exp2(2u*log2e)+1)
  float e = fast_exp2(u * 2.8853900817779268f);
  return 1.0f - 2.0f * fast_rcp(e + 1.0f);
#endif
}

// ---------------------------------------------------------------------------
// Setup kernel: Wvo[e] = Wo[e] * Wv[e]   (attn = Xn * Wvo^T)
// ---------------------------------------------------------------------------
__global__ __launch_bounds__(256) void build_wvo_kernel(
    const float* __restrict__ Wv, const float* __restrict__ Wo,
    float* __restrict__ Wvo) {
  int e = blockIdx.x;
  int t = threadIdx.x;            // 256 threads: j = out row, d = out col
  int j = t >> 4, d = t & 15;
  const float* wo = Wo + e * 256;
  const float* wv = Wv + e * 256;
  float s = 0.0f;
#pragma unroll
  for (int v = 0; v < 16; ++v) s = fmaf(wo[j * 16 + v], wv[v * 16 + d], s);
  Wvo[e * 256 + t] = s;
}

// ---------------------------------------------------------------------------
// Main kernel: one wave32 per 16-row tile.
// ---------------------------------------------------------------------------
__global__ __launch_bounds__(32) void moe_block_kernel(
    const float* __restrict__ state, const float* __restrict__ Wvo,
    const float* __restrict__ W1, const float* __restrict__ b1,
    const float* __restrict__ W2, const float* __restrict__ b2,
    float* __restrict__ out) {
  __shared__ float sSt[16 * 17];   // raw state tile        (row-major, pad 17)
  __shared__ float sXn[16 * 17];   // layernorm(state)
  __shared__ float sX1[16 * 17];   // state + attn
  __shared__ float sXn2[16 * 17];  // layernorm(x1)
  __shared__ float sH[16 * 68];    // silu hidden 16x64     (pad 68 -> no conflicts)
  __shared__ float sOpc[16];       // per-row opcode

  const int lane = threadIdx.x & 31;
  const int ln = lane & 15;   // column / row-in-half
  const int hi = lane >> 4;   // half-wave id
  const int row0 = blockIdx.x * 16;

  // --- Stage 0: load 16 rows, layernorm(state) -> LDS -----------------------
  if (hi == 0) {
    const float* sr = state + (size_t)(row0 + ln) * ND;
    float x[16];
    const float4* sr4 = reinterpret_cast<const float4*>(sr);
#pragma unroll
    for (int q = 0; q < 4; ++q) {
      float4 v = sr4[q];
      x[4 * q + 0] = v.x; x[4 * q + 1] = v.y; x[4 * q + 2] = v.z; x[4 * q + 3] = v.w;
    }
#pragma unroll
    for (int c = 0; c < 16; ++c) sSt[ln * 17 + c] = x[c];
    sOpc[ln] = x[OPCODE_COL];
    float m = 0.0f;
#pragma unroll
    for (int c = 0; c < 16; ++c) m += x[c];
    m *= (1.0f / 16.0f);
    float var = 0.0f;
#pragma unroll
    for (int c = 0; c < 16; ++c) { float t = x[c] - m; var = fmaf(t, t, var); }
    var *= (1.0f / 16.0f);
    float inv = fast_rsq(var + LN_EPS);          // v_rsq_f32
#pragma unroll
    for (int c = 0; c < 16; ++c) sXn[ln * 17 + c] = (x[c] - m) * inv;
  }
  __syncthreads();

  // Expert-invariant preloads.
  v2f axn[4];
#pragma unroll
  for (int k = 0; k < 4; ++k) axn[k] = frag16(sXn, ln, hi, 4 * k, 17);
  float gti[8], stcd[8];
#pragma unroll
  for (int i = 0; i < 8; ++i) {
    gti[i]  = sOpc[i + 8 * hi] * L2E20;         // opcode * 20*log2e (gate base)
    stcd[i] = sSt[(i + 8 * hi) * 17 + ln];      // state[m][ln] in C/D layout
  }

  v8f outAcc = {};

  for (int e = 0; e < NE; ++e) {
    const float se = (float)e * L2E20;

    // --- attn = Xn * Wvo[e]^T  (4 WMMAs; B-frags batched) ------------------
    const float* wvo = Wvo + e * 256;
    v2f bvo[4];
#pragma unroll
    for (int k = 0; k < 4; ++k) bvo[k] = frag16(wvo, ln, hi, 4 * k, ND);
    v8f attn = {};
#pragma unroll
    for (int k = 0; k < 4; ++k) attn = wmma4(axn[k], bvo[k], attn);

    // x1 = state + attn  -> LDS (C/D layout: row m = i+8*hi, col ln)
#pragma unroll
    for (int i = 0; i < 8; ++i)
      sX1[(i + 8 * hi) * 17 + ln] = stcd[i] + attn[i];
    __syncthreads();

    // --- layernorm(x1) -> sXn2 ---------------------------------------------
    if (hi == 0) {
      float r[16];
#pragma unroll
      for (int c = 0; c < 16; ++c) r[c] = sX1[ln * 17 + c];
      float m = 0.0f;
#pragma unroll
      for (int c = 0; c < 16; ++c) m += r[c];
      m *= (1.0f / 16.0f);
      float var = 0.0f;
#pragma unroll
      for (int c = 0; c < 16; ++c) { float t = r[c] - m; var = fmaf(t, t, var); }
      var *= (1.0f / 16.0f);
      float inv = fast_rsq(var + LN_EPS);
#pragma unroll
      for (int c = 0; c < 16; ++c) sXn2[ln * 17 + c] = (r[c] - m) * inv;
    }
    __syncthreads();

    v2f a2[4];
#pragma unroll
    for (int k = 0; k < 4; ++k) a2[k] = frag16(sXn2, ln, hi, 4 * k, 17);

    // --- h = silu(Xn2 * W1[e]^T + b1[e])  (16 WMMAs, 4 accs) -> sH ---------
    const float* w1e = W1 + (size_t)e * NDFF * ND;
    const float* b1e = b1 + (size_t)e * NDFF;
    float hb[4];
#pragma unroll
    for (int nt = 0; nt < 4; ++nt) hb[nt] = 0.5f * b1e[nt * 16 + ln];
    v2f bw[16];
#pragma unroll
    for (int nt = 0; nt < 4; ++nt)
#pragma unroll
      for (int k = 0; k < 4; ++k)
        bw[nt * 4 + k] = frag16(w1e + nt * 16 * ND, ln, hi, 4 * k, ND);
    v8f hacc[4];
#pragma unroll
    for (int nt = 0; nt < 4; ++nt) hacc[nt] = (v8f){};
#pragma unroll
    for (int k = 0; k < 4; ++k)          // round-robin: no same-acc RAW chains
#pragma unroll
      for (int nt = 0; nt < 4; ++nt)
        hacc[nt] = wmma4(a2[k], bw[nt * 4 + k], hacc[nt]);
#pragma unroll
    for (int nt = 0; nt < 4; ++nt) {
      float u[8], t[8];
#pragma unroll
      for (int i = 0; i < 8; ++i) u[i] = fmaf(0.5f, hacc[nt][i], hb[nt]);  // z/2
#pragma unroll
      for (int i = 0; i < 8; ++i) t[i] = fast_tanh(u[i]);
#pragma unroll
      for (int i = 0; i < 8; ++i)        // silu(z) = u + u*tanh(u)
        sH[(i + 8 * hi) * 68 + nt * 16 + ln] = fmaf(u[i], t[i], u[i]);
    }
    __syncthreads();

    // --- ffn = H * W2[e]^T  (K=64, 16 WMMAs; 2 batches of 8) ---------------
    v8f f = {};
    const float* w2e = W2 + (size_t)e * ND * NDFF;
#pragma unroll
    for (int half = 0; half < 2; ++half) {
      v2f ah[8], bh[8];
#pragma unroll
      for (int k = 0; k < 8; ++k) {
        int k0 = 32 * half + 4 * k;
        ah[k] = frag16(sH, ln, hi, k0, 68);
        bh[k] = frag16(w2e, ln, hi, k0, NDFF);
      }
#pragma unroll
      for (int k = 0; k < 8; ++k) f = wmma4(ah[k], bh[k], f);
    }

    // --- x2 = x1 + ffn + b2 ; out += gate(e) * x2 --------------------------
    // gate = sig(a)sig(b) = 1/(1 + e^-a + e^-b + e^-20), and 1+e^-20 == 1.0f
    // in fp32; with u = (opc-e)*20*log2e: g = rcp(1 + 2^(-u-c) + 2^(u-c)).
    const float bias2 = b2[(size_t)e * ND + ln];
    float x2v[8], g[8];
#pragma unroll
    for (int i = 0; i < 8; ++i)
      x2v[i] = sX1[(i + 8 * hi) * 17 + ln] + f[i] + bias2;
#pragma unroll
    for (int i = 0; i < 8; ++i) {
      float u = gti[i] - se;
      float e1 = fast_exp2(-u - L2E10);
      float e2 = fast_exp2(u - L2E10);
      g[i] = fast_rcp(1.0f + e1 + e2);
    }
#pragma unroll
    for (int i = 0; i < 8; ++i) outAcc[i] = fmaf(g[i], x2v[i], outAcc[i]);
    __syncthreads();  // protect sX1/sXn2/sH before next expert overwrites
  }

  // --- store (row m = i+8*hi, col ln) --------------------------------------
#pragma unroll
  for (int i = 0; i < 8; ++i)
    out[(size_t)(row0 + i + 8 * hi) * ND + ln] = outAcc[i];
}

// ---------------------------------------------------------------------------
extern "C" void kernel_launch(void* const* d_in, const int* in_sizes, int n_in,
                              void* d_out, int out_size, void* d_ws, size_t ws_size,
                              hipStream_t stream) {
  const float* state = (const float*)d_in[0];
  // d_in[1] = Wq, d_in[2] = Wk : mathematically dead (softmax over size-1 axis)
  const float* Wv = (const float*)d_in[3];
  const float* Wo = (const float*)d_in[4];
  const float* W1 = (const float*)d_in[5];
  const float* b1 = (const float*)d_in[6];
  const float* W2 = (const float*)d_in[7];
  const float* b2 = (const float*)d_in[8];
  float* out = (float*)d_out;
  float* Wvo = (float*)d_ws;  // 39*256*4 = 39936 bytes of scratch

  build_wvo_kernel<<<NE, 256, 0, stream>>>(Wv, Wo, Wvo);
  moe_block_kernel<<<NB / 16, 32, 0, stream>>>(state, Wvo, W1, b1, W2, b2, out);
  (void)in_sizes; (void)n_in; (void)out_size; (void)ws_size;
}